// ChamferLoss_53661321396251
// MI455X (gfx1250) — compile-verified
//
#include <hip/hip_runtime.h>
#include <hip/hip_bf16.h>

typedef __attribute__((ext_vector_type(2))) float v2f;
typedef __attribute__((ext_vector_type(8))) float v8f;

__global__ void chamfer_zero_out(float* out) {
    if (threadIdx.x == 0 && blockIdx.x == 0) out[0] = 0.0f;
}

// One direction of Chamfer: for each point p in P (B x Np x 3), min squared
// distance to any q in Q (B x Nq x 3); per-batch mean over Np, scaled by
// `scale` (= 1/(B*Np)), accumulated into out[0].
//
// Each wave owns a 16-row tile of P and sweeps all 16-col tiles of Q with
// V_WMMA_F32_16X16X4_F32 (K=4, D=3 padded with zero).
__global__ __launch_bounds__(256)
void chamfer_dir_kernel(const float* __restrict__ P, const float* __restrict__ Q,
                        float* __restrict__ out, int Np, int Nq, float scale)
{
    const int lane   = threadIdx.x & 31;
    const int wave   = (blockIdx.x * blockDim.x + threadIdx.x) >> 5;
    const int tilesPerBatch = Np >> 4;
    const int b = wave / tilesPerBatch;
    const int t = wave % tilesPerBatch;

    const float* Pb = P + (size_t)b * Np * 3;
    const float* Qb = Q + (size_t)b * Nq * 3;

    const int laneHi = lane >> 4;   // 0: lanes 0-15 (K0/K1, rows 0-7 of C)
                                    // 1: lanes 16-31 (K2/K3, rows 8-15 of C)
    const int lane16 = lane & 15;

    // ---- A tile: 16 P points. 32-bit A 16x4 layout:
    // lanes 0-15: VGPR0=K0, VGPR1=K1 ; lanes 16-31: VGPR0=K2, VGPR1=K3(=0 pad)
    const float* pa = Pb + (size_t)(t * 16 + lane16) * 3;
    const float pa0 = pa[0], pa1 = pa[1], pa2 = pa[2];
    v2f A;
    A.x = laneHi ? pa2 : pa0;
    A.y = laneHi ? 0.0f : pa1;

    // Row squared-norms matching the C/D layout: VGPR r holds row r + 8*laneHi.
    float x2row[8];
    #pragma unroll
    for (int r = 0; r < 8; ++r) {
        const float* pr = Pb + (size_t)(t * 16 + r + 8 * laneHi) * 3;
        const float a0 = pr[0], a1 = pr[1], a2 = pr[2];
        x2row[r] = a0 * a0 + a1 * a1 + a2 * a2;
    }

    float rowmin[8];
    #pragma unroll
    for (int r = 0; r < 8; ++r) rowmin[r] = 3.4e38f;

    const int qTiles = Nq >> 4;
    for (int qt = 0; qt < qTiles; ++qt) {
        // B tile: 16 Q points, 32-bit B 4x16 layout mirrors A (lane = col).
        const float* pq = Qb + (size_t)(qt * 16 + lane16) * 3;
        const float q0 = pq[0], q1 = pq[1], q2 = pq[2];
        const float y2 = q0 * q0 + q1 * q1 + q2 * q2;  // col norm (per lane16)
        v2f Bm;
        Bm.x = laneHi ? q2 : q0;
        Bm.y = laneHi ? 0.0f : q1;

        v8f C = {};
        // D = A x B (+0): xy[row, col] tile
        C = __builtin_amdgcn_wmma_f32_16x16x4_f32(
            /*neg_a=*/false, A, /*neg_b=*/false, Bm,
            /*c_mod=*/(short)0, C, /*reuse_a=*/false, /*reuse_b=*/false);

        #pragma unroll
        for (int r = 0; r < 8; ++r) {
            float d2 = fmaf(-2.0f, C[r], x2row[r] + y2);
            d2 = fmaxf(d2, 0.0f);
            rowmin[r] = fminf(rowmin[r], d2);
        }
    }

    // Min over columns: reduce across the 16 lanes of each half-wave.
    #pragma unroll
    for (int m = 8; m >= 1; m >>= 1) {
        #pragma unroll
        for (int r = 0; r < 8; ++r) {
            const float o = __shfl_xor(rowmin[r], m, 32);
            rowmin[r] = fminf(rowmin[r], o);
        }
    }

    // Sum the 8 row-mins held by this half, then add the other half's 8 rows.
    float s = 0.0f;
    #pragma unroll
    for (int r = 0; r < 8; ++r) s += rowmin[r];
    s += __shfl_xor(s, 16, 32);

    if (lane == 0) atomicAdd(out, s * scale);
}

extern "C" void kernel_launch(void* const* d_in, const int* in_sizes, int n_in,
                              void* d_out, int out_size, void* d_ws, size_t ws_size,
                              hipStream_t stream) {
    const float* x = (const float*)d_in[0];
    const float* y = (const float*)d_in[1];
    float* out = (float*)d_out;

    const int B = 4;                       // from reference setup
    const int N = in_sizes[0] / (B * 3);   // 8192
    const int M = in_sizes[1] / (B * 3);   // 8192

    chamfer_zero_out<<<1, 1, 0, stream>>>(out);

    // Direction x -> y: min over y for each x, mean over N.
    {
        const int waves   = B * (N / 16);          // 2048
        const int threads = waves * 32;            // 65536
        const int blocks  = threads / 256;         // exact, all waves full
        chamfer_dir_kernel<<<blocks, 256, 0, stream>>>(x, y, out, N, M,
                                                       1.0f / (float)(B * N));
    }
    // Direction y -> x: min over x for each y, mean over M.
    {
        const int waves   = B * (M / 16);
        const int threads = waves * 32;
        const int blocks  = threads / 256;
        chamfer_dir_kernel<<<blocks, 256, 0, stream>>>(y, x, out, M, N,
                                                       1.0f / (float)(B * M));
    }
}